// GDN_51135880626555
// MI455X (gfx1250) — compile-verified
//
#include <hip/hip_runtime.h>
#include <math.h>

#define BB 64
#define NN 1024
#define D_IN 128
#define DD 64
#define TOPK 20
#define BN (BB*NN)
#define EPSV 1e-5f
#define NEG_SLOPE 0.2f

typedef float v2f __attribute__((ext_vector_type(2)));
typedef float v8f __attribute__((ext_vector_type(8)));

__device__ __forceinline__ float waveRedSum(float v) {
    #pragma unroll
    for (int m = 16; m > 0; m >>= 1) v += __shfl_xor(v, m, 32);
    return v;
}
__device__ __forceinline__ float waveRedMax(float v) {
    #pragma unroll
    for (int m = 16; m > 0; m >>= 1) v = fmaxf(v, __shfl_xor(v, m, 32));
    return v;
}

// ---------------- K0: embedding row norms ----------------
__global__ void k_norms(const float* __restrict__ emb, float* __restrict__ nrm) {
    int j = blockIdx.x * blockDim.x + threadIdx.x;
    if (j < NN) {
        float s = 0.f;
        for (int c = 0; c < DD; ++c) { float v = emb[j * DD + c]; s += v * v; }
        nrm[j] = sqrtf(s);
    }
}

// ---------------- K1: cosine sim row + top-20 (lowest-index tie break) ----------------
__global__ void k_topk(const float* __restrict__ emb, const float* __restrict__ nrm,
                       int* __restrict__ topk) {
    __shared__ float wi[DD];
    __shared__ float vals[NN];
    __shared__ float rv[256];
    __shared__ int   rj[256];
    int i = blockIdx.x;
    int t = threadIdx.x;
    if (t < DD) wi[t] = emb[i * DD + t];
    __syncthreads();
    float inv_ni = 1.0f / nrm[i];
    for (int j = t; j < NN; j += 256) {
        float s = 0.f;
        for (int c = 0; c < DD; ++c) s += wi[c] * emb[j * DD + c];
        vals[j] = s * inv_ni / nrm[j];
    }
    __syncthreads();
    for (int r = 0; r < TOPK; ++r) {
        float bv = -INFINITY; int bj = 0;
        for (int j = t; j < NN; j += 256) {
            float v = vals[j];
            if (v > bv || (v == bv && j < bj)) { bv = v; bj = j; }
        }
        rv[t] = bv; rj[t] = bj;
        __syncthreads();
        for (int s = 128; s > 0; s >>= 1) {
            if (t < s) {
                float ov = rv[t + s]; int oj = rj[t + s];
                if (ov > rv[t] || (ov == rv[t] && oj < rj[t])) { rv[t] = ov; rj[t] = oj; }
            }
            __syncthreads();
        }
        if (t == 0) { topk[i * TOPK + r] = rj[0]; vals[rj[0]] = -INFINITY; }
        __syncthreads();
    }
}

// ---------------- K2: g = x @ lin_w via V_WMMA_F32_16X16X4_F32 ----------------
// Block = 128 threads = 4 waves; wave w owns output tile [row0:row0+16, 16w:16w+16].
// A 16x4 layout: lanes 0-15 hold (K=0,K=1) for M=lane; lanes 16-31 hold (K=2,K=3).
// B 4x16 layout mirrors the K split: VGPR0 = K0/K2 rows, VGPR1 = K1/K3 rows, N = lane%16.
__global__ void k_gemm_wmma(const float* __restrict__ x, const float* __restrict__ lw,
                            float* __restrict__ g) {
    int wave = threadIdx.x >> 5;
    int lane = threadIdx.x & 31;
    int row0 = blockIdx.x * 16;
    int col0 = wave * 16;
    int lhalf = lane >> 4;            // 0: K pair (0,1); 1: K pair (2,3)
    int l = lane & 15;

    v8f acc = {0.f, 0.f, 0.f, 0.f, 0.f, 0.f, 0.f, 0.f};
    const float* arow = x + (size_t)(row0 + l) * D_IN + (lhalf ? 2 : 0);
    const float* bcol = lw + col0 + l;

    #pragma unroll 4
    for (int k = 0; k < D_IN; k += 4) {
        v2f a = *(const v2f*)(arow + k);          // contiguous (K, K+1) pair
        int kb = k + (lhalf ? 2 : 0);
        v2f b;
        b.x = bcol[(size_t)kb * DD];
        b.y = bcol[(size_t)(kb + 1) * DD];
        acc = __builtin_amdgcn_wmma_f32_16x16x4_f32(
            /*neg_a=*/false, a, /*neg_b=*/false, b,
            /*c_mod=*/(short)0, acc, /*reuse_a=*/false, /*reuse_b=*/false);
    }

    // D layout: VGPR r -> M = r (lanes 0-15) or r+8 (lanes 16-31), N = lane%16
    int ncol = col0 + l;
    int rbase = row0 + (lhalf ? 8 : 0);
    #pragma unroll
    for (int r = 0; r < 8; ++r)
        g[(size_t)(rbase + r) * DD + ncol] = acc[r];
}

// ---------------- K3: per-node attention scores (wave per node) ----------------
__global__ void k_scores(const float* __restrict__ g, const float* __restrict__ emb,
                         const float* __restrict__ att_i, const float* __restrict__ att_j,
                         const float* __restrict__ att_em_i, const float* __restrict__ att_em_j,
                         float* __restrict__ s_src, float* __restrict__ s_dst) {
    int wave = threadIdx.x >> 5, lane = threadIdx.x & 31;
    int n = blockIdx.x * 8 + wave;
    int i = n & (NN - 1);
    float g0 = g[(size_t)n * DD + lane], g1 = g[(size_t)n * DD + lane + 32];
    float e0 = emb[i * DD + lane],       e1 = emb[i * DD + lane + 32];
    float ss = g0 * att_j[lane] + g1 * att_j[lane + 32]
             + e0 * att_em_j[lane] + e1 * att_em_j[lane + 32];
    float sd = g0 * att_i[lane] + g1 * att_i[lane + 32]
             + e0 * att_em_i[lane] + e1 * att_em_i[lane + 32];
    ss = waveRedSum(ss);
    sd = waveRedSum(sd);
    if (lane == 0) { s_src[n] = ss; s_dst[n] = sd; }
}

// ---------------- K4: softmax over 21 incoming edges + weighted aggregation ----------------
__global__ void k_attn_agg(const float* __restrict__ g, const float* __restrict__ s_src,
                           const float* __restrict__ s_dst, const int* __restrict__ topk,
                           const float* __restrict__ bias, float* __restrict__ agg) {
    int wave = threadIdx.x >> 5, lane = threadIdx.x & 31;
    int n = blockIdx.x * 8 + wave;
    int i = n & (NN - 1);
    int b = n >> 10;
    float sd = s_dst[n];
    int srcn = n;
    float a = -INFINITY;
    if (lane < TOPK) {
        int j = topk[i * TOPK + lane];
        srcn = b * NN + j;
        a = s_src[srcn] + sd;
        a = a > 0.f ? a : NEG_SLOPE * a;                 // leaky_relu
        if (j == i) a = -1e30f;                          // masked (removed self edge)
    } else if (lane == TOPK) {                           // appended self loop
        a = s_src[n] + sd;
        a = a > 0.f ? a : NEG_SLOPE * a;
    }
    float m = waveRedMax(a);
    float ew = (lane <= TOPK) ? expf(a - m) : 0.f;
    float denom = waveRedSum(ew);
    float acc0 = 0.f, acc1 = 0.f;
    #pragma unroll
    for (int e = 0; e <= TOPK; ++e) {
        float we = __shfl(ew, e, 32);
        int sn = __shfl(srcn, e, 32);
        acc0 += we * g[(size_t)sn * DD + lane];
        acc1 += we * g[(size_t)sn * DD + lane + 32];
    }
    float inv = 1.0f / denom;
    agg[(size_t)n * DD + lane]      = acc0 * inv + bias[lane];
    agg[(size_t)n * DD + lane + 32] = acc1 * inv + bias[lane + 32];
}

// ---------------- K5: per-channel partial sum/sumsq over 65536 rows ----------------
__global__ void k_stats(const float* __restrict__ src, float* __restrict__ partials) {
    __shared__ float sb[256];
    __shared__ float sb2[256];
    int t = threadIdx.x;
    float s = 0.f, s2 = 0.f;
    size_t base = (size_t)blockIdx.x * 256 + t;
    #pragma unroll 4
    for (int it = 0; it < 64; ++it) {
        float v = src[base + (size_t)it * 65536];  // stride multiple of 64 -> channel fixed
        s += v; s2 += v * v;
    }
    sb[t] = s; sb2[t] = s2;
    __syncthreads();
    if (t < 64) {
        s  = sb[t]  + sb[t + 64]  + sb[t + 128]  + sb[t + 192];
        s2 = sb2[t] + sb2[t + 64] + sb2[t + 128] + sb2[t + 192];
        partials[blockIdx.x * 128 + t]      = s;
        partials[blockIdx.x * 128 + 64 + t] = s2;
    }
}

__global__ void k_bn_finalize(const float* __restrict__ partials,
                              const float* __restrict__ gamma, const float* __restrict__ beta,
                              float* __restrict__ scale, float* __restrict__ shift) {
    int c = threadIdx.x;
    if (c < DD) {
        float s = 0.f, s2 = 0.f;
        for (int p = 0; p < 256; ++p) { s += partials[p * 128 + c]; s2 += partials[p * 128 + 64 + c]; }
        float mean = s / (float)BN;
        float var = s2 / (float)BN - mean * mean;      // biased variance (torch training)
        float sc = gamma[c] * rsqrtf(var + EPSV);
        scale[c] = sc;
        shift[c] = beta[c] - mean * sc;
    }
}

// ---------------- K6: h2 = relu(bn1(agg)) * embedding ----------------
__global__ void k_h2(const float* __restrict__ agg, const float* __restrict__ scale1,
                     const float* __restrict__ shift1, const float* __restrict__ emb,
                     float* __restrict__ h2) {
    int t = threadIdx.x; int c = t & 63;
    float sc = scale1[c], sh = shift1[c];
    size_t base = (size_t)blockIdx.x * 256 + t;
    #pragma unroll 4
    for (int it = 0; it < 64; ++it) {
        size_t idx = base + (size_t)it * 65536;
        int row = (int)(idx >> 6);
        int i = row & (NN - 1);
        float h = agg[idx] * sc + sh;
        h = h > 0.f ? h : 0.f;
        h2[idx] = h * emb[i * DD + c];
    }
}

// ---------------- K8: out = relu(bn2(h2)) @ w_out + b_out (wave per node) ----------------
__global__ void k_out(const float* __restrict__ h2, const float* __restrict__ scale2,
                      const float* __restrict__ shift2, const float* __restrict__ w_out,
                      const float* __restrict__ b_out, float* __restrict__ out) {
    int wave = threadIdx.x >> 5, lane = threadIdx.x & 31;
    int n = blockIdx.x * 8 + wave;
    float v0 = h2[(size_t)n * DD + lane] * scale2[lane] + shift2[lane];
    float v1 = h2[(size_t)n * DD + lane + 32] * scale2[lane + 32] + shift2[lane + 32];
    v0 = v0 > 0.f ? v0 : 0.f;
    v1 = v1 > 0.f ? v1 : 0.f;
    float s = v0 * w_out[lane] + v1 * w_out[lane + 32];
    s = waveRedSum(s);
    if (lane == 0) out[n] = s + b_out[0];
}

extern "C" void kernel_launch(void* const* d_in, const int* in_sizes, int n_in,
                              void* d_out, int out_size, void* d_ws, size_t ws_size,
                              hipStream_t stream) {
    const float* x        = (const float*)d_in[0];   // data  [B,N,D_IN]
    // d_in[1] = org_edge_index: unused by the reference computation
    const float* emb      = (const float*)d_in[2];   // embedding [N,D]
    const float* lw       = (const float*)d_in[3];   // lin_w [D_IN,D]
    const float* att_i    = (const float*)d_in[4];
    const float* att_j    = (const float*)d_in[5];
    const float* att_em_i = (const float*)d_in[6];
    const float* att_em_j = (const float*)d_in[7];
    const float* gnn_bias = (const float*)d_in[8];
    const float* bn1_g    = (const float*)d_in[9];
    const float* bn1_b    = (const float*)d_in[10];
    const float* bn2_g    = (const float*)d_in[11];
    const float* bn2_b    = (const float*)d_in[12];
    const float* w_out    = (const float*)d_in[13];
    const float* b_out    = (const float*)d_in[14];
    float* out = (float*)d_out;

    // Workspace layout (bytes)
    char* ws = (char*)d_ws;
    float* g      = (float*)(ws + 0);                     // 16 MB
    float* agg    = (float*)(ws + (size_t)16777216);      // 16 MB
    float* h2     = (float*)(ws + (size_t)33554432);      // 16 MB
    float* s_src  = (float*)(ws + (size_t)50331648);      // 256 KB
    float* s_dst  = (float*)(ws + (size_t)50593792);      // 256 KB
    float* nrm    = (float*)(ws + (size_t)50855936);      // 4 KB
    int*   topk   = (int*)  (ws + (size_t)50860032);      // 80 KB
    float* part1  = (float*)(ws + (size_t)50941952);      // 128 KB
    float* part2  = (float*)(ws + (size_t)51073024);      // 128 KB
    float* scale1 = (float*)(ws + (size_t)51204096);
    float* shift1 = (float*)(ws + (size_t)51204352);
    float* scale2 = (float*)(ws + (size_t)51204608);
    float* shift2 = (float*)(ws + (size_t)51204864);

    k_norms<<<4, 256, 0, stream>>>(emb, nrm);
    k_topk<<<NN, 256, 0, stream>>>(emb, nrm, topk);
    k_gemm_wmma<<<BN / 16, 128, 0, stream>>>(x, lw, g);
    k_scores<<<BN / 8, 256, 0, stream>>>(g, emb, att_i, att_j, att_em_i, att_em_j, s_src, s_dst);
    k_attn_agg<<<BN / 8, 256, 0, stream>>>(g, s_src, s_dst, topk, gnn_bias, agg);
    k_stats<<<256, 256, 0, stream>>>(agg, part1);
    k_bn_finalize<<<1, 64, 0, stream>>>(part1, bn1_g, bn1_b, scale1, shift1);
    k_h2<<<256, 256, 0, stream>>>(agg, scale1, shift1, emb, h2);
    k_stats<<<256, 256, 0, stream>>>(h2, part2);
    k_bn_finalize<<<1, 64, 0, stream>>>(part2, bn2_g, bn2_b, scale2, shift2);
    k_out<<<BN / 8, 256, 0, stream>>>(h2, scale2, shift2, w_out, b_out, out);
    (void)in_sizes; (void)n_in; (void)out_size; (void)ws_size;
}